// GCNEncoder_47528108098278
// MI455X (gfx1250) — compile-verified
//
#include <hip/hip_runtime.h>

#define IN_DIM 128
#define HID    64
#define EMB    128

typedef float v2f __attribute__((ext_vector_type(2)));
typedef float v8f __attribute__((ext_vector_type(8)));

// ---------------- degree / normalization ----------------

__global__ void k_init_deg(float* deg, int n) {
  int i = blockIdx.x * blockDim.x + threadIdx.x;
  if (i < n) deg[i] = 1.0f;  // self-loop weight
}

__global__ void k_deg_scatter(const long long* __restrict__ dst,
                              const float* __restrict__ ew,
                              float* deg, int nE) {
  int e = blockIdx.x * blockDim.x + threadIdx.x;
  if (e < nE) atomicAdd(&deg[dst[e]], ew[e]);
}

__global__ void k_rsqrt_inplace(float* deg, int n) {
  int i = blockIdx.x * blockDim.x + threadIdx.x;
  if (i < n) {
    float d = deg[i];
    deg[i] = (d > 0.0f) ? rsqrtf(d) : 0.0f;
  }
}

// ---------------- fp32 WMMA GEMM: C[M,N] = A[M,K] @ B[K,N] ----------------
// One wave per 16x16 output tile. Uses V_WMMA_F32_16X16X4_F32.
// A 16x4 layout : lanes 0-15 hold M=lane, VGPR0/1 = K=(0,1); lanes 16-31 K=(2,3)
// B 4x16 layout : lanes 0-15 hold N=lane, VGPR0/1 = K=(0,1); lanes 16-31 K=(2,3)
// C/D 16x16     : VGPR r -> M=r (lanes 0-15) / M=r+8 (lanes 16-31), N=lane&15

template <int K, int N>
__global__ void k_gemm_wmma(const float* __restrict__ A,
                            const float* __restrict__ B,
                            float* __restrict__ C, int M) {
  const int ntiles = N / 16;
  int wave = (int)((blockIdx.x * (unsigned)blockDim.x + threadIdx.x) >> 5);
  int lane = threadIdx.x & 31;
  int mt = wave / ntiles;
  int nt = wave - mt * ntiles;
  if (mt * 16 >= M) return;  // wave-uniform exit: EXEC all-ones at WMMA

  int r  = lane & 15;   // row of A / col of B-C
  int kh = lane >> 4;   // K half-select

  const float* Ap = A + (size_t)(mt * 16 + r) * K + 2 * kh;
  const float* Bp = B + (size_t)(2 * kh) * N + nt * 16 + r;

  v8f acc = {};
#pragma unroll 4
  for (int k0 = 0; k0 < K; k0 += 4) {
    v2f a, b;
    a.x = Ap[k0];
    a.y = Ap[k0 + 1];
    b.x = Bp[(size_t)k0 * N];
    b.y = Bp[(size_t)k0 * N + N];
    acc = __builtin_amdgcn_wmma_f32_16x16x4_f32(
        /*neg_a=*/false, a, /*neg_b=*/false, b,
        /*c_mod=*/(short)0, acc, /*reuse_a=*/false, /*reuse_b=*/false);
  }

  float* Cp = C + (size_t)(mt * 16 + kh * 8) * N + nt * 16 + r;
#pragma unroll
  for (int i = 0; i < 8; ++i) Cp[(size_t)i * N] = acc[i];
}

// ---------------- edge scatter: out[dst] += h[src] * (dis[s]*w*dis[d]) ----

template <int D>
__global__ void k_edge_scatter(const float* __restrict__ h,
                               const long long* __restrict__ src,
                               const long long* __restrict__ dst,
                               const float* __restrict__ ew,
                               const float* __restrict__ dis,
                               float* __restrict__ out, int nE) {
  const int G = D / 4;
  long long total = (long long)nE * G;
  for (long long idx = blockIdx.x * (long long)blockDim.x + threadIdx.x;
       idx < total; idx += (long long)gridDim.x * blockDim.x) {
    int e = (int)(idx / G);
    int g = (int)(idx - (long long)e * G);
    long long s = src[e], d = dst[e];
    float nrm = dis[s] * ew[e] * dis[d];
    const float4 hv = *(const float4*)(h + (size_t)s * D + g * 4);
    float* o = out + (size_t)d * D + g * 4;
    atomicAdd(o + 0, hv.x * nrm);
    atomicAdd(o + 1, hv.y * nrm);
    atomicAdd(o + 2, hv.z * nrm);
    atomicAdd(o + 3, hv.w * nrm);
  }
}

// ---- finalize: out = relu(out + h*dis^2 + bias)  (self-loop + bias fused) --

__global__ void k_finalize(const float* __restrict__ h,
                           const float* __restrict__ dis,
                           const float* __restrict__ bias,
                           float* __restrict__ out, long long total, int D) {
  long long idx = blockIdx.x * (long long)blockDim.x + threadIdx.x;
  if (idx < total) {
    int i = (int)(idx / D);
    int d = (int)(idx - (long long)i * D);
    float di = dis[i];
    float v = out[idx] + h[idx] * di * di + bias[d];
    out[idx] = v > 0.0f ? v : 0.0f;
  }
}

// ---------------- global mean pool ----------------

__global__ void k_pool_scatter(const float* __restrict__ h,
                               const long long* __restrict__ batch,
                               float* __restrict__ sums,
                               float* __restrict__ cnt, int n) {
  int idx = blockIdx.x * blockDim.x + threadIdx.x;  // (node, 4-dim group)
  int node = idx >> 5;   // EMB/4 = 32 groups per node
  int g = idx & 31;
  if (node < n) {
    long long b = batch[node];
    const float4 hv = *(const float4*)(h + (size_t)node * EMB + g * 4);
    float* o = sums + (size_t)b * EMB + g * 4;
    atomicAdd(o + 0, hv.x);
    atomicAdd(o + 1, hv.y);
    atomicAdd(o + 2, hv.z);
    atomicAdd(o + 3, hv.w);
    if (g == 0) atomicAdd(&cnt[b], 1.0f);
  }
}

__global__ void k_pool_div(const float* __restrict__ sums,
                           const float* __restrict__ cnt,
                           float* __restrict__ out, int total) {
  int i = blockIdx.x * blockDim.x + threadIdx.x;
  if (i < total) out[i] = sums[i] / fmaxf(cnt[i / EMB], 1.0f);
}

// ---------------- launcher ----------------

extern "C" void kernel_launch(void* const* d_in, const int* in_sizes, int n_in,
                              void* d_out, int out_size, void* d_ws, size_t ws_size,
                              hipStream_t stream) {
  (void)n_in; (void)ws_size;
  const float*     x     = (const float*)d_in[0];
  const long long* ei    = (const long long*)d_in[1];
  const float*     ew    = (const float*)d_in[2];
  const long long* batch = (const long long*)d_in[3];
  const float*     W1    = (const float*)d_in[4];
  const float*     b1    = (const float*)d_in[5];
  const float*     W2    = (const float*)d_in[6];
  const float*     b2    = (const float*)d_in[7];
  float* out = (float*)d_out;

  const int N = in_sizes[0] / IN_DIM;   // 50000
  const int E = in_sizes[1] / 2;        // 800000
  const int G = out_size / EMB;         // 256
  const long long* src = ei;
  const long long* dstp = ei + E;

  // workspace layout (floats), 16B-aligned chunks
  float* ws = (float*)d_ws;
  size_t o = 0;
  float* deg = ws + o; o += ((size_t)N + 127) & ~(size_t)127;  // deg -> dis in place
  float* h1  = ws + o; o += (size_t)N * HID;
  float* a1  = ws + o; o += (size_t)N * HID;
  float* h2  = ws + o; o += (size_t)N * EMB;
  float* a2  = ws + o; o += (size_t)N * EMB;
  float* sums = ws + o; o += (size_t)G * EMB;
  float* cnt  = ws + o; o += (size_t)G;

  const int T = 256;  // 8 waves per block (wave32)
  auto blk = [](long long n, int t) { return (unsigned)((n + t - 1) / t); };

  // 1) symmetric normalization coefficients
  k_init_deg<<<blk(N, T), T, 0, stream>>>(deg, N);
  k_deg_scatter<<<blk(E, T), T, 0, stream>>>(dstp, ew, deg, E);
  k_rsqrt_inplace<<<blk(N, T), T, 0, stream>>>(deg, N);

  // 2) layer 1: h1 = x @ W1  (WMMA fp32)
  {
    long long waves = (long long)((N + 15) / 16) * (HID / 16);
    k_gemm_wmma<IN_DIM, HID><<<blk(waves * 32, T), T, 0, stream>>>(x, W1, h1, N);
  }
  hipMemsetAsync(a1, 0, (size_t)N * HID * sizeof(float), stream);
  k_edge_scatter<HID><<<blk((long long)E * (HID / 4), T), T, 0, stream>>>(
      h1, src, dstp, ew, deg, a1, E);
  k_finalize<<<blk((long long)N * HID, T), T, 0, stream>>>(
      h1, deg, b1, a1, (long long)N * HID, HID);

  // 3) layer 2: h2 = a1 @ W2  (WMMA fp32)
  {
    long long waves = (long long)((N + 15) / 16) * (EMB / 16);
    k_gemm_wmma<HID, EMB><<<blk(waves * 32, T), T, 0, stream>>>(a1, W2, h2, N);
  }
  hipMemsetAsync(a2, 0, (size_t)N * EMB * sizeof(float), stream);
  k_edge_scatter<EMB><<<blk((long long)E * (EMB / 4), T), T, 0, stream>>>(
      h2, src, dstp, ew, deg, a2, E);
  k_finalize<<<blk((long long)N * EMB, T), T, 0, stream>>>(
      h2, deg, b2, a2, (long long)N * EMB, EMB);

  // 4) global mean pool
  hipMemsetAsync(sums, 0, (size_t)G * EMB * sizeof(float), stream);
  hipMemsetAsync(cnt, 0, (size_t)G * sizeof(float), stream);
  k_pool_scatter<<<blk((long long)N * 32, T), T, 0, stream>>>(a2, batch, sums, cnt, N);
  k_pool_div<<<blk(out_size, T), T, 0, stream>>>(sums, cnt, out, out_size);
}